// MessagePassing_18098992185815
// MI455X (gfx1250) — compile-verified
//
#include <hip/hip_runtime.h>
#include <stdint.h>

#define WPB   8            // waves per block (wave32)
#define TPB   (WPB * 32)
#define CHUNK 32           // edges per wave-chunk

struct __align__(16) EdgeTile {
    unsigned long long src[CHUNK];   // 256 B
    unsigned long long dst[CHUNK];   // 256 B
    float              w[CHUNK];     // 128 B
};                                   // 640 B per tile

// ---- CDNA5 async global->LDS staging (ASYNCcnt path) -----------------------
__device__ __forceinline__ void async_tile_issue(unsigned lds_src, unsigned lds_dst,
                                                 unsigned lds_w,
                                                 unsigned off_src, unsigned off_dst,
                                                 unsigned off_w,
                                                 const void* eidx_base,
                                                 const void* w_base) {
    // Make sure prior DS reads of the buffer being overwritten have drained.
    asm volatile("s_wait_dscnt 0" ::: "memory");
    // GVS mode: global address is the 64-bit base in saddr plus the 32-bit offset in vaddr; data lands at the LDS address in vdst
    asm volatile("global_load_async_to_lds_b64 %0, %1, %2"
                 :: "v"(lds_src), "v"(off_src), "s"(eidx_base) : "memory");
    asm volatile("global_load_async_to_lds_b64 %0, %1, %2"
                 :: "v"(lds_dst), "v"(off_dst), "s"(eidx_base) : "memory");
    asm volatile("global_load_async_to_lds_b32 %0, %1, %2"
                 :: "v"(lds_w), "v"(off_w), "s"(w_base) : "memory");
}

__device__ __forceinline__ void process_tile(const EdgeTile& t,
                                             const float* __restrict__ x,
                                             float* __restrict__ out,
                                             int cnt, int lane) {
    // Prefetch the gathered x rows for this tile (one row per lane; 2x128B lines).
    {
        size_t s = (size_t)t.src[lane];
        const float* row = x + s * 64;
        __builtin_prefetch(row, 0, 3);
        __builtin_prefetch(row + 32, 0, 3);
    }
    if (cnt == 32) {
#pragma unroll 8
        for (int j = 0; j < 32; ++j) {
            size_t s   = (size_t)t.src[j];            // broadcast DS read
            size_t d   = (size_t)t.dst[j];
            float  wgt = t.w[j];
            const float2 xv = *reinterpret_cast<const float2*>(x + s * 64 + lane * 2);
            float* op = out + d * 64 + lane * 2;
            unsafeAtomicAdd(op,     xv.x * wgt);      // global_atomic_add_f32, no return
            unsafeAtomicAdd(op + 1, xv.y * wgt);
        }
    } else {
        for (int j = 0; j < cnt; ++j) {
            size_t s   = (size_t)t.src[j];
            size_t d   = (size_t)t.dst[j];
            float  wgt = t.w[j];
            const float2 xv = *reinterpret_cast<const float2*>(x + s * 64 + lane * 2);
            float* op = out + d * 64 + lane * 2;
            unsafeAtomicAdd(op,     xv.x * wgt);
            unsafeAtomicAdd(op + 1, xv.y * wgt);
        }
    }
}

__global__ void __launch_bounds__(TPB)
scatter_add_kernel(const float* __restrict__ x,
                   const float* __restrict__ ew,
                   const long long* __restrict__ eidx,   // [2, E]: src then dst
                   float* __restrict__ out,
                   int E) {
    __shared__ EdgeTile tiles[WPB][2];

    const int lane    = threadIdx.x & 31;
    const int wv      = threadIdx.x >> 5;
    const int wave_id = blockIdx.x * WPB + wv;
    const int nwaves  = gridDim.x * WPB;
    const int nchunks = (E + CHUNK - 1) / CHUNK;

    // Low 32 bits of a generic pointer to LDS are the LDS byte offset.
    unsigned lsrc[2], ldst[2], lw[2];
#pragma unroll
    for (int b = 0; b < 2; ++b) {
        lsrc[b] = (unsigned)(uintptr_t)&tiles[wv][b].src[lane];
        ldst[b] = (unsigned)(uintptr_t)&tiles[wv][b].dst[lane];
        lw[b]   = (unsigned)(uintptr_t)&tiles[wv][b].w[lane];
    }

    auto issue = [&](int chunk, int b) {
        int e = chunk * CHUNK + lane;
        if (e >= E) e = E - 1;                      // clamp tail (safe read)
        unsigned off_src = (unsigned)e * 8u;
        unsigned off_dst = (unsigned)(E + e) * 8u;  // dst row of edge_index
        unsigned off_w   = (unsigned)e * 4u;
        async_tile_issue(lsrc[b], ldst[b], lw[b], off_src, off_dst, off_w,
                         (const void*)eidx, (const void*)ew);
    };

    int cur = wave_id;
    int buf = 0;
    if (cur < nchunks) issue(cur, buf);

    for (; cur < nchunks; cur += nwaves) {
        int nxt = cur + nwaves;
        bool has_next = nxt < nchunks;
        if (has_next) {
            issue(nxt, buf ^ 1);
            asm volatile("s_wait_asynccnt 3" ::: "memory");  // chunk `cur` landed
        } else {
            asm volatile("s_wait_asynccnt 0" ::: "memory");
        }
        int cnt = E - cur * CHUNK;
        if (cnt > CHUNK) cnt = CHUNK;
        process_tile(tiles[wv][buf], x, out, cnt, lane);
        buf ^= 1;
    }
}

// Zero the output AND warm the (same-shaped) x array into the 192MB L2 so the
// scatter phase's random gathers + atomics all hit L2 instead of HBM.
__global__ void zero_and_warm_kernel(float4* __restrict__ o4, int n4,
                                     float* __restrict__ o, int n,
                                     const float* __restrict__ x) {
    int i = blockIdx.x * blockDim.x + threadIdx.x;
    if (i < n4) {
        o4[i] = float4{0.f, 0.f, 0.f, 0.f};
        if ((i & 7) == 0) {
            // one global_prefetch per 128B line of x (same extent as out)
            __builtin_prefetch(x + (size_t)i * 4, 0, 3);
        }
    }
    int t = n4 * 4 + i;
    if (t < n) o[t] = 0.f;
}

extern "C" void kernel_launch(void* const* d_in, const int* in_sizes, int n_in,
                              void* d_out, int out_size, void* d_ws, size_t ws_size,
                              hipStream_t stream) {
    const float*     x    = (const float*)d_in[0];       // [N, 64] f32
    const float*     ew   = (const float*)d_in[1];       // [E] f32
    const long long* eidx = (const long long*)d_in[2];   // [2, E] i64
    float*           out  = (float*)d_out;               // [N, 64] f32
    const int E = in_sizes[1];

    // 1) zero the (poisoned) output + warm x into L2
    int n4 = out_size / 4;
    int zb = (n4 + 255) / 256;
    if (zb < 1) zb = 1;
    zero_and_warm_kernel<<<zb, 256, 0, stream>>>((float4*)out, n4, out, out_size, x);

    // 2) async-staged, double-buffered scatter-add
    int nchunks = (E + CHUNK - 1) / CHUNK;
    int blocks  = (nchunks + WPB - 1) / WPB;
    if (blocks > 2048) blocks = 2048;
    if (blocks < 1) blocks = 1;
    scatter_add_kernel<<<blocks, TPB, 0, stream>>>(x, ew, eidx, out, E);
}